// HybridInference_11390253269311
// MI455X (gfx1250) — compile-verified
//
#include <hip/hip_runtime.h>

// ---------------------------------------------------------------------------
// HybridInference on MI455X (gfx1250): fp32 WMMA (V_WMMA_F32_16X16X4_F32)
// ---------------------------------------------------------------------------

typedef float v2f __attribute__((ext_vector_type(2)));
typedef float v8f __attribute__((ext_vector_type(8)));

#define BDIM 16
#define TDIM 1000
#define XD 8
#define YD 4
#define HD 48
#define ED 104        // 2*HD + XD
#define GD 144        // 3*HD
#define NT 63         // ceil(TDIM/16)
#define GAMMA_F 1e-4f
#define ITERS_H 200

__device__ __forceinline__ v8f wmma4(v2f a, v2f b, v8f c) {
  // D = A(16x4,f32) * B(4x16,f32) + C(16x16,f32)
  return __builtin_amdgcn_wmma_f32_16x16x4_f32(false, a, false, b, (short)0, c,
                                               false, false);
}

// ---------------------------------------------------------------------------
// Prep: pack a weight matrix (O x I, row-major) into WMMA A-fragment order.
// frag[((r*KT + kk)*32 + lane)*2 + v] = W[r*16 + lane%16][kk*4 + v + 2*(lane/16)]
// Rows beyond O are zero-padded (needed for W_dec, O=8).
// ---------------------------------------------------------------------------
__global__ void pack_weights(const float* __restrict__ W, float* __restrict__ dst,
                             int O, int I) {
  int RT = (O + 15) / 16;
  int KT = I / 4;
  int total = RT * KT * 64;
  int stride = gridDim.x * blockDim.x;
  for (int idx = blockIdx.x * blockDim.x + threadIdx.x; idx < total; idx += stride) {
    int v = idx & 1;
    int lane = (idx >> 1) & 31;
    int fk = idx >> 6;
    int kk = fk % KT;
    int r = fk / KT;
    int m = lane & 15;
    int hi = lane >> 4;
    int row = r * 16 + m;
    int col = kk * 4 + v + 2 * hi;
    dst[idx] = (row < O) ? W[row * I + col] : 0.0f;
  }
}

// ---------------------------------------------------------------------------
// Prep: invert Q (8x8) and R (4x4) (SPD, Gauss-Jordan) and build the small
// message matrices. Output layout (floats):
//   [0..63]  NQi = -inv(Q)
//   [64..]   QiF = inv(Q) @ F
//   [128..]  FtQi = F^T @ inv(Q)
//   [192..]  FtQiF = F^T @ inv(Q) @ F
//   [256..287] HtRi = H^T @ inv(R)      (8x4, idx x*4+y)
//   [288..351] HtRiH = H^T @ inv(R) @ H (8x8)
// ---------------------------------------------------------------------------
__global__ void prep_small(const float* __restrict__ F, const float* __restrict__ Hm,
                           const float* __restrict__ Q, const float* __restrict__ R,
                           float* __restrict__ sm) {
  if (threadIdx.x != 0 || blockIdx.x != 0) return;

  float M[8][16];
  float Qi[8][8], Ri[4][4];

  // invert Q
  for (int i = 0; i < 8; ++i)
    for (int j = 0; j < 8; ++j) {
      M[i][j] = Q[i * 8 + j];
      M[i][8 + j] = (i == j) ? 1.0f : 0.0f;
    }
  for (int c = 0; c < 8; ++c) {
    float ip = 1.0f / M[c][c];
    for (int j = 0; j < 16; ++j) M[c][j] *= ip;
    for (int r = 0; r < 8; ++r) {
      if (r == c) continue;
      float f = M[r][c];
      for (int j = 0; j < 16; ++j) M[r][j] -= f * M[c][j];
    }
  }
  for (int i = 0; i < 8; ++i)
    for (int j = 0; j < 8; ++j) Qi[i][j] = M[i][8 + j];

  // invert R
  float N_[4][8];
  for (int i = 0; i < 4; ++i)
    for (int j = 0; j < 4; ++j) {
      N_[i][j] = R[i * 4 + j];
      N_[i][4 + j] = (i == j) ? 1.0f : 0.0f;
    }
  for (int c = 0; c < 4; ++c) {
    float ip = 1.0f / N_[c][c];
    for (int j = 0; j < 8; ++j) N_[c][j] *= ip;
    for (int r = 0; r < 4; ++r) {
      if (r == c) continue;
      float f = N_[r][c];
      for (int j = 0; j < 8; ++j) N_[r][j] -= f * N_[c][j];
    }
  }
  for (int i = 0; i < 4; ++i)
    for (int j = 0; j < 4; ++j) Ri[i][j] = N_[i][4 + j];

  float QiF[8][8], FtQi[8][8], HtRi[8][4];
  for (int i = 0; i < 8; ++i)
    for (int j = 0; j < 8; ++j) {
      float s1 = 0.0f, s2 = 0.0f;
      for (int k = 0; k < 8; ++k) {
        s1 += Qi[i][k] * F[k * 8 + j];   // Qi @ F
        s2 += F[k * 8 + i] * Qi[k][j];   // F^T @ Qi
      }
      QiF[i][j] = s1;
      FtQi[i][j] = s2;
    }
  for (int x = 0; x < 8; ++x)
    for (int y = 0; y < 4; ++y) {
      float s = 0.0f;
      for (int z = 0; z < 4; ++z) s += Hm[z * XD + x] * Ri[z][y];
      HtRi[x][y] = s;
    }

  for (int i = 0; i < 8; ++i)
    for (int j = 0; j < 8; ++j) {
      sm[i * 8 + j] = -Qi[i][j];
      sm[64 + i * 8 + j] = QiF[i][j];
      sm[128 + i * 8 + j] = FtQi[i][j];
      float s = 0.0f;
      for (int k = 0; k < 8; ++k) s += FtQi[i][k] * F[k * 8 + j];
      sm[192 + i * 8 + j] = s;
      float s2 = 0.0f;
      for (int y = 0; y < 4; ++y) s2 += HtRi[i][y] * Hm[y * XD + j];
      sm[288 + i * 8 + j] = s2;
    }
  for (int x = 0; x < 8; ++x)
    for (int y = 0; y < 4; ++y) sm[256 + x * 4 + y] = HtRi[x][y];
}

// ---------------------------------------------------------------------------
// Prep: xs0 = H^T y,  hy = W_hy @ y^T + b_hy,  hxA = hx0
// ---------------------------------------------------------------------------
__global__ void init_state(const float* __restrict__ ys, const float* __restrict__ hx0,
                           const float* __restrict__ Hm, const float* __restrict__ Why,
                           const float* __restrict__ bhy,
                           float* __restrict__ xs0, float* __restrict__ hxA,
                           float* __restrict__ hyb) {
  int stride = gridDim.x * blockDim.x;
  int tid = blockIdx.x * blockDim.x + threadIdx.x;
  for (int i = tid; i < BDIM * XD * TDIM; i += stride) {
    int t = i % TDIM;
    int x = (i / TDIM) % XD;
    int b = i / (TDIM * XD);
    float s = 0.0f;
#pragma unroll
    for (int y = 0; y < YD; ++y) s += Hm[y * XD + x] * ys[((size_t)b * TDIM + t) * YD + y];
    xs0[i] = s;
  }
  for (int i = tid; i < BDIM * HD * TDIM; i += stride) {
    int t = i % TDIM;
    int h = (i / TDIM) % HD;
    int b = i / (TDIM * HD);
    float s = bhy[h];
#pragma unroll
    for (int y = 0; y < YD; ++y) s += Why[h * YD + y] * ys[((size_t)b * TDIM + t) * YD + y];
    hyb[i] = s;
    hxA[i] = hx0[i];
  }
}

// ---------------------------------------------------------------------------
// Main per-iteration kernel: one wave per (b, 16-column t-tile).
// ---------------------------------------------------------------------------
struct IterArgs {
  const float* xs_in;
  float* xs_out;
  const float* hx_in;
  float* hx_out;
  const float* ys;
  const float* hy;
  const float* sm;
  const float* fW1p; const float* fW1f; const float* fW1y;
  const float* fW2p; const float* fW2f; const float* fW2y;
  const float* fWih; const float* fWhh; const float* fWdec;
  const float* b1p;  const float* b1f;  const float* b1y;
  const float* b2p;  const float* b2f;  const float* b2y;
  const float* bih;  const float* bhh;  const float* bdec;
  float* out_pred;
  float* out_final;   // null except last iteration
};

__global__ __launch_bounds__(32) void iter_kernel(IterArgs A) {
  __shared__ float s_msg[3][XD * 16];   // mp, mf, my
  __shared__ float s_mid[HD * 16];      // edge hidden layer
  __shared__ float s_agg[HD * 16];      // summed edge outputs
  __shared__ float s_gx[GD * 16];       // W_ih @ agg + b_ih
  __shared__ float s_gh[GD * 16];       // W_hh @ hx  + b_hh
  __shared__ float s_hn[HD * 16];       // new hidden state

  const int tile = blockIdx.x;
  const int b = tile / NT;
  const int tBase = (tile % NT) * 16;
  const int lane = threadIdx.x;
  const int n = lane & 15;
  const int hi = lane >> 4;
  const int t = tBase + n;
  const int tc = t < TDIM ? t : TDIM - 1;
  const bool valid = (t < TDIM);

  const float* hxb = A.hx_in + (size_t)b * HD * TDIM;

  // ----- messages (per-lane mat-vec over the 8x8 precomputed matrices) -----
  {
    float xc[XD], xp[XD], xf[XD], yc[YD];
#pragma unroll
    for (int x = 0; x < XD; ++x) {
      const float* px = A.xs_in + (size_t)(b * XD + x) * TDIM;
      xc[x] = px[tc];
      xp[x] = (t > 0) ? px[tc - 1] : 0.0f;
      xf[x] = (t < TDIM - 1) ? px[tc + 1] : 0.0f;
    }
#pragma unroll
    for (int y = 0; y < YD; ++y) yc[y] = A.ys[((size_t)b * TDIM + tc) * YD + y];

    const float* NQi = A.sm;
    const float* QiF = A.sm + 64;
    const float* FtQi = A.sm + 128;
    const float* FtQiF = A.sm + 192;
    const float* HtRi = A.sm + 256;
    const float* HtRiH = A.sm + 288;

#pragma unroll
    for (int jj = 0; jj < 4; ++jj) {
      int r = hi * 4 + jj;
      float mp = 0.0f, mf = 0.0f, my = 0.0f;
#pragma unroll
      for (int x = 0; x < XD; ++x) {
        mp += NQi[r * 8 + x] * xc[x] + QiF[r * 8 + x] * xp[x];
        mf += FtQi[r * 8 + x] * xf[x] - FtQiF[r * 8 + x] * xc[x];
        my -= HtRiH[r * 8 + x] * xc[x];
      }
#pragma unroll
      for (int y = 0; y < YD; ++y) my += HtRi[r * 4 + y] * yc[y];
      if (t == 0) mp = 0.0f;
      if (t >= TDIM - 1) mf = 0.0f;
      s_msg[0][r * 16 + n] = mp;
      s_msg[1][r * 16 + n] = mf;
      s_msg[2][r * 16 + n] = my;
    }
  }
  __syncthreads();

  // ----- three edge MLPs, accumulating agg in registers -----
  const float* fW1[3] = {A.fW1p, A.fW1f, A.fW1y};
  const float* fW2[3] = {A.fW2p, A.fW2f, A.fW2y};
  const float* b1[3] = {A.b1p, A.b1f, A.b1y};

  v8f accAgg[3];
#pragma unroll
  for (int r = 0; r < 3; ++r)
#pragma unroll
    for (int v = 0; v < 8; ++v) {
      int row = r * 16 + v + 8 * hi;
      accAgg[r][v] = A.b2p[row] + A.b2f[row] + A.b2y[row];
    }

  for (int e = 0; e < 3; ++e) {
    const v2f* wf = (const v2f*)fW1[e];
    // layer 1: mid = relu(W1 @ [h; hn; m] + b1)
    for (int r = 0; r < 3; ++r) {
      v8f acc;
#pragma unroll
      for (int v = 0; v < 8; ++v) acc[v] = b1[e][r * 16 + v + 8 * hi];
#pragma unroll
      for (int kk = 0; kk < 26; ++kk) {
        v2f a = wf[(r * 26 + kk) * 32 + lane];
        v2f bf;
#pragma unroll
        for (int v = 0; v < 2; ++v) {
          int row = kk * 4 + v + 2 * hi;
          float val;
          if (row < HD) {
            val = hxb[(size_t)row * TDIM + tc];
          } else if (row < 2 * HD) {
            int rr = row - HD;
            if (e == 0)
              val = (t > 0) ? hxb[(size_t)rr * TDIM + tc - 1] : 0.0f;
            else if (e == 1)
              val = (t < TDIM - 1) ? hxb[(size_t)rr * TDIM + tc + 1] : 0.0f;
            else
              val = A.hy[((size_t)b * HD + rr) * TDIM + tc];
          } else {
            val = s_msg[e][(row - 2 * HD) * 16 + n];
          }
          bf[v] = val;
        }
        acc = wmma4(a, bf, acc);
      }
#pragma unroll
      for (int v = 0; v < 8; ++v)
        s_mid[(r * 16 + v + 8 * hi) * 16 + n] = fmaxf(acc[v], 0.0f);
    }
    __syncthreads();

    // layer 2: agg += W2 @ mid
    const v2f* wf2 = (const v2f*)fW2[e];
    for (int r = 0; r < 3; ++r) {
#pragma unroll
      for (int kk = 0; kk < 12; ++kk) {
        v2f a = wf2[(r * 12 + kk) * 32 + lane];
        v2f bf;
#pragma unroll
        for (int v = 0; v < 2; ++v) bf[v] = s_mid[(kk * 4 + v + 2 * hi) * 16 + n];
        accAgg[r] = wmma4(a, bf, accAgg[r]);
      }
    }
    __syncthreads();
  }

  // agg to LDS for re-fragmentation
#pragma unroll
  for (int r = 0; r < 3; ++r)
#pragma unroll
    for (int v = 0; v < 8; ++v)
      s_agg[(r * 16 + v + 8 * hi) * 16 + n] = accAgg[r][v];
  __syncthreads();

  // ----- GRU gates: gx = W_ih@agg + b_ih, gh = W_hh@hx + b_hh -----
  {
    const v2f* wih = (const v2f*)A.fWih;
    const v2f* whh = (const v2f*)A.fWhh;
    for (int r = 0; r < 9; ++r) {
      v8f ax, ah;
#pragma unroll
      for (int v = 0; v < 8; ++v) {
        int row = r * 16 + v + 8 * hi;
        ax[v] = A.bih[row];
        ah[v] = A.bhh[row];
      }
#pragma unroll
      for (int kk = 0; kk < 12; ++kk) {
        v2f a1 = wih[(r * 12 + kk) * 32 + lane];
        v2f a2 = whh[(r * 12 + kk) * 32 + lane];
        v2f bfa, bfh;
#pragma unroll
        for (int v = 0; v < 2; ++v) {
          int row = kk * 4 + v + 2 * hi;
          bfa[v] = s_agg[row * 16 + n];
          bfh[v] = hxb[(size_t)row * TDIM + tc];
        }
        ax = wmma4(a1, bfa, ax);
        ah = wmma4(a2, bfh, ah);
      }
#pragma unroll
      for (int v = 0; v < 8; ++v) {
        s_gx[(r * 16 + v + 8 * hi) * 16 + n] = ax[v];
        s_gh[(r * 16 + v + 8 * hi) * 16 + n] = ah[v];
      }
    }
  }
  __syncthreads();

  // ----- GRU elementwise update -----
#pragma unroll
  for (int k = 0; k < 24; ++k) {
    int j = lane + k * 32;             // 768 = 48 rows * 16 cols
    int row = j >> 4;
    int col = j & 15;
    int tcol = tBase + col;
    int tcc = tcol < TDIM ? tcol : TDIM - 1;
    float xr = s_gx[row * 16 + col];
    float xz = s_gx[(row + HD) * 16 + col];
    float xn = s_gx[(row + 2 * HD) * 16 + col];
    float hr = s_gh[row * 16 + col];
    float hz = s_gh[(row + HD) * 16 + col];
    float hnm = s_gh[(row + 2 * HD) * 16 + col];
    float hold = hxb[(size_t)row * TDIM + tcc];
    float rg = 1.0f / (1.0f + __expf(-(xr + hr)));
    float zg = 1.0f / (1.0f + __expf(-(xz + hz)));
    float ng = tanhf(xn + rg * hnm);
    float hnew = (1.0f - zg) * ng + zg * hold;
    s_hn[row * 16 + col] = hnew;
    if (tcol < TDIM) A.hx_out[((size_t)b * HD + row) * TDIM + tcol] = hnew;
  }
  __syncthreads();

  // ----- eps = W_dec @ hnew + b_dec ; xs update -----
  {
    const v2f* wdec = (const v2f*)A.fWdec;
    v8f acc;
#pragma unroll
    for (int v = 0; v < 8; ++v) {
      int m = v + 8 * hi;
      acc[v] = (m < XD) ? A.bdec[m] : 0.0f;
    }
#pragma unroll
    for (int kk = 0; kk < 12; ++kk) {
      v2f a = wdec[kk * 32 + lane];
      v2f bf;
#pragma unroll
      for (int v = 0; v < 2; ++v) bf[v] = s_hn[(kk * 4 + v + 2 * hi) * 16 + n];
      acc = wmma4(a, bf, acc);
    }
    if (hi == 0 && valid) {
#pragma unroll
      for (int v = 0; v < 8; ++v) {
        float msum = s_msg[0][v * 16 + n] + s_msg[1][v * 16 + n] + s_msg[2][v * 16 + n];
        size_t oi = ((size_t)b * XD + v) * TDIM + t;
        float xnew = A.xs_in[oi] + GAMMA_F * (acc[v] + msum);
        A.xs_out[oi] = xnew;
        A.out_pred[oi] = xnew;
        if (A.out_final) A.out_final[oi] = xnew;
      }
    }
  }
}

// ---------------------------------------------------------------------------
extern "C" void kernel_launch(void* const* d_in, const int* in_sizes, int n_in,
                              void* d_out, int out_size, void* d_ws, size_t ws_size,
                              hipStream_t stream) {
  const float* ys = (const float*)d_in[0];
  const float* hx0 = (const float*)d_in[1];
  const float* F = (const float*)d_in[2];
  const float* Hm = (const float*)d_in[3];
  const float* Q = (const float*)d_in[4];
  const float* R = (const float*)d_in[5];
  const float* W_hy = (const float*)d_in[6];
  const float* b_hy = (const float*)d_in[7];
  const float* W1p = (const float*)d_in[8];
  const float* b1p = (const float*)d_in[9];
  const float* W2p = (const float*)d_in[10];
  const float* b2p = (const float*)d_in[11];
  const float* W1f = (const float*)d_in[12];
  const float* b1f = (const float*)d_in[13];
  const float* W2f = (const float*)d_in[14];
  const float* b2f = (const float*)d_in[15];
  const float* W1y = (const float*)d_in[16];
  const float* b1y = (const float*)d_in[17];
  const float* W2y = (const float*)d_in[18];
  const float* b2y = (const float*)d_in[19];
  const float* Wih = (const float*)d_in[20];
  const float* bih = (const float*)d_in[21];
  const float* Whh = (const float*)d_in[22];
  const float* bhh = (const float*)d_in[23];
  const float* Wdec = (const float*)d_in[24];
  const float* bdec = (const float*)d_in[25];

  float* ws = (float*)d_ws;
  // packed weight fragment sizes
  const int P_W1 = 3 * 26 * 64;   // 4992
  const int P_W2 = 3 * 12 * 64;   // 2304
  const int P_G = 9 * 12 * 64;    // 6912
  const int P_D = 1 * 12 * 64;    // 768
  float* fW1p = ws;
  float* fW1f = fW1p + P_W1;
  float* fW1y = fW1f + P_W1;
  float* fW2p = fW1y + P_W1;
  float* fW2f = fW2p + P_W2;
  float* fW2y = fW2f + P_W2;
  float* fWih = fW2y + P_W2;
  float* fWhh = fWih + P_G;
  float* fWdec = fWhh + P_G;
  float* sm = fWdec + P_D;
  float* xsA = sm + 352;
  float* xsB = xsA + BDIM * XD * TDIM;
  float* hxA = xsB + BDIM * XD * TDIM;
  float* hxB = hxA + BDIM * HD * TDIM;
  float* hyb = hxB + BDIM * HD * TDIM;

  pack_weights<<<28, 256, 0, stream>>>(W1p, fW1p, HD, ED);
  pack_weights<<<28, 256, 0, stream>>>(W1f, fW1f, HD, ED);
  pack_weights<<<28, 256, 0, stream>>>(W1y, fW1y, HD, ED);
  pack_weights<<<28, 256, 0, stream>>>(W2p, fW2p, HD, HD);
  pack_weights<<<28, 256, 0, stream>>>(W2f, fW2f, HD, HD);
  pack_weights<<<28, 256, 0, stream>>>(W2y, fW2y, HD, HD);
  pack_weights<<<28, 256, 0, stream>>>(Wih, fWih, GD, HD);
  pack_weights<<<28, 256, 0, stream>>>(Whh, fWhh, GD, HD);
  pack_weights<<<28, 256, 0, stream>>>(Wdec, fWdec, XD, HD);
  prep_small<<<1, 32, 0, stream>>>(F, Hm, Q, R, sm);
  init_state<<<1024, 256, 0, stream>>>(ys, hx0, Hm, W_hy, b_hy, xsA, hxA, hyb);

  float* out = (float*)d_out;
  const size_t XSN = (size_t)BDIM * XD * TDIM;  // 128000

  float* xs_in = xsA;
  float* xs_out = xsB;
  float* hx_in = hxA;
  float* hx_out = hxB;

  for (int it = 0; it < ITERS_H; ++it) {
    IterArgs a;
    a.xs_in = xs_in; a.xs_out = xs_out;
    a.hx_in = hx_in; a.hx_out = hx_out;
    a.ys = ys; a.hy = hyb; a.sm = sm;
    a.fW1p = fW1p; a.fW1f = fW1f; a.fW1y = fW1y;
    a.fW2p = fW2p; a.fW2f = fW2f; a.fW2y = fW2y;
    a.fWih = fWih; a.fWhh = fWhh; a.fWdec = fWdec;
    a.b1p = b1p; a.b1f = b1f; a.b1y = b1y;
    a.b2p = b2p; a.b2f = b2f; a.b2y = b2y;
    a.bih = bih; a.bhh = bhh; a.bdec = bdec;
    a.out_pred = out + XSN + (size_t)it * XSN;
    a.out_final = (it == ITERS_H - 1) ? out : nullptr;
    iter_kernel<<<BDIM * NT, 32, 0, stream>>>(a);

    float* tmp = xs_in; xs_in = xs_out; xs_out = tmp;
    tmp = hx_in; hx_in = hx_out; hx_out = tmp;
  }
}